// TSKModel_50749333569915
// MI455X (gfx1250) — compile-verified
//
#include <hip/hip_runtime.h>
#include <hip/hip_bf16.h>

typedef __attribute__((ext_vector_type(2))) float v2f;
typedef __attribute__((ext_vector_type(4))) float v4f;
typedef __attribute__((ext_vector_type(8))) float v8f;

#define NROWS 32768
#define DIMS  64
#define RULES 32
#define D1    65          // 1 + DIMS
#define EPSF  1e-12f
#define TM    64          // rows per block
#define XS_STRIDE 68      // LDS row stride (floats): avoids bank conflicts, keeps 16B align

__global__ __launch_bounds__(256) void tsk_wmma_kernel(
    const float* __restrict__ Xz, const float* __restrict__ centers,
    const float* __restrict__ sigmas, const float* __restrict__ theta,
    float* __restrict__ y_out, float* __restrict__ w_out, float* __restrict__ phi_out)
{
    __shared__ __align__(16) float Xs[TM * XS_STRIDE];     // 17.0 KB  X tile
    __shared__ v2f Ba2[(DIMS / 2) * RULES];                // 8 KB  a   = 1/(s+e)^2, [kpair][rule]
    __shared__ v2f Bb2[(DIMS / 2) * RULES];                // 8 KB  -2*c*a
    __shared__ v2f Bt2[(DIMS / 2) * RULES];                // 8 KB  theta[:,1:]
    __shared__ __align__(16) float w_s[TM * RULES];        // 8 KB  log_w -> w
    __shared__ float g_s[TM * RULES];                      // 8 KB  rule outputs
    __shared__ float kr_s[RULES];                          // sum c^2*a
    __shared__ float bias_s[RULES];                        // theta[:,0]

    const int tid  = threadIdx.x;
    const int row0 = blockIdx.x * TM;

    // ---- Stage 0a: per-rule constant matrices into LDS (B-operand layout) ----
    for (int idx = tid; idx < RULES * DIMS; idx += 256) {
        int r = idx >> 6;           // rule
        int d = idx & 63;           // feature
        float s   = sigmas[r * DIMS + d] + EPSF;
        float inv = 1.0f / s;
        float a   = inv * inv;
        float c   = centers[r * DIMS + d];
        int kp = d >> 1, comp = d & 1;
        ((float*)Ba2)[(kp * RULES + r) * 2 + comp] = a;
        ((float*)Bb2)[(kp * RULES + r) * 2 + comp] = -2.0f * c * a;
        ((float*)Bt2)[(kp * RULES + r) * 2 + comp] = theta[r * D1 + 1 + d];
    }
    if (tid < RULES) {
        float acc = 0.0f;
        for (int d = 0; d < DIMS; ++d) {
            float s   = sigmas[tid * DIMS + d] + EPSF;
            float inv = 1.0f / s;
            float c   = centers[tid * DIMS + d];
            acc += c * c * inv * inv;
        }
        kr_s[tid]   = acc;
        bias_s[tid] = theta[tid * D1];
    }
    // ---- Stage 0b: X tile (64x64 f32) -> LDS, b128 coalesced ----
    {
        const v4f* Xv = (const v4f*)(Xz + (size_t)row0 * DIMS);
        for (int idx = tid; idx < TM * (DIMS / 4); idx += 256) {
            int r  = idx >> 4;      // 16 float4 per row
            int c4 = idx & 15;
            v4f v = Xv[idx];        // rows contiguous in memory
            *(v4f*)&Xs[r * XS_STRIDE + c4 * 4] = v;
        }
    }
    __syncthreads();

    // ---- Stage 1: WMMA GEMMs. 8 waves = 4 row-slabs x 2 rule-tiles ----
    const int lane = tid & 31;
    const int wv   = tid >> 5;
    const int slab = wv >> 1;        // 16-row slab within tile
    const int tt   = wv & 1;         // rule half (0..15 / 16..31)
    const int m    = lane & 15;      // row (A) / rule (B,C,D) index within tile
    const int hw   = lane >> 4;      // half-wave selects K+2 (A/B), M+8 (C/D)
    const int kh   = hw << 1;
    const int colg = tt * 16 + m;    // global rule index for this lane
    {
        v8f acc  = {0.f,0.f,0.f,0.f,0.f,0.f,0.f,0.f};  // x^2*a - 2x*c*a
        v8f accg = {0.f,0.f,0.f,0.f,0.f,0.f,0.f,0.f};  // theta1 . x
        const float* xrow = &Xs[(slab * 16 + m) * XS_STRIDE];
        #pragma unroll
        for (int k0 = 0; k0 < DIMS; k0 += 4) {
            int k  = k0 + kh;
            v2f x  = *(const v2f*)&xrow[k];
            v2f xsq; xsq[0] = x[0] * x[0]; xsq[1] = x[1] * x[1];
            int kp = k >> 1;
            v2f bA = Ba2[kp * RULES + colg];
            v2f bB = Bb2[kp * RULES + colg];
            v2f bT = Bt2[kp * RULES + colg];
            acc  = __builtin_amdgcn_wmma_f32_16x16x4_f32(false, xsq, false, bA, (short)0, acc,  false, false);
            acc  = __builtin_amdgcn_wmma_f32_16x16x4_f32(false, x,   false, bB, (short)0, acc,  false, false);
            accg = __builtin_amdgcn_wmma_f32_16x16x4_f32(false, x,   false, bT, (short)0, accg, false, false);
        }
        float krv = kr_s[colg];
        float bv  = bias_s[colg];
        #pragma unroll
        for (int i = 0; i < 8; ++i) {       // C/D layout: row = i + 8*hw, col = m
            int rr = slab * 16 + i + 8 * hw;
            w_s[rr * RULES + colg] = -0.5f * (acc[i] + krv);   // log_w
            g_s[rr * RULES + colg] = accg[i] + bv;             // g = theta0 + theta1.x
        }
    }
    __syncthreads();

    // ---- Stage 2: row softmax (with reference epsilon) + y = sum_r w*g ----
    if (tid < TM) {
        float l[RULES];
        float mx = -3.4e38f;
        #pragma unroll
        for (int r = 0; r < RULES; ++r) { l[r] = w_s[tid * RULES + r]; mx = fmaxf(mx, l[r]); }
        float sum = 0.0f;
        #pragma unroll
        for (int r = 0; r < RULES; ++r) { float e = __expf(l[r] - mx); l[r] = e; sum += e; }
        float inv = 1.0f / (sum + EPSF);
        float y = 0.0f;
        #pragma unroll
        for (int r = 0; r < RULES; ++r) {
            float wn = l[r] * inv;
            w_s[tid * RULES + r] = wn;
            y += wn * g_s[tid * RULES + r];
        }
        y_out[row0 + tid] = y;
    }
    __syncthreads();

    // ---- Stage 3: streaming outputs (store-bandwidth bound: ~277 MB total) ----
    // w: coalesced b128 non-temporal
    {
        const v4f* ws4  = (const v4f*)w_s;
        v4f* wout4 = (v4f*)(w_out + (size_t)row0 * RULES);
        for (int idx = tid; idx < (TM * RULES) / 4; idx += 256)
            __builtin_nontemporal_store(ws4[idx], &wout4[idx]);
    }

    // Phi: each wave streams 8 rows; 512 B per wave-instruction via b128 NT stores.
    // Row base is 16B-aligned (2080*4 bytes/row), c = 4q keeps float4 aligned.
    for (int rloc = wv * 8; rloc < wv * 8 + 8; ++rloc) {
        const float* wr = &w_s[rloc * RULES];
        const float* xr = &Xs[rloc * XS_STRIDE];
        float* dst = phi_out + (size_t)(row0 + rloc) * (RULES * D1);
        for (int q = lane; q < (RULES * D1) / 4; q += 32) {   // 520 float4 per row
            int c = q * 4;
            v4f v;
            #pragma unroll
            for (int i = 0; i < 4; ++i) {
                int ci = c + i;
                int r  = ci / D1;                 // rule index (mul-hi magic)
                int j  = ci - r * D1;             // 0 -> w_r, else w_r * x[j-1]
                int jm = j - 1; if (jm < 0) jm = 0;
                float xv = xr[jm];
                v[i] = (j == 0) ? wr[r] : wr[r] * xv;
            }
            __builtin_nontemporal_store(v, (v4f*)(dst + c));
        }
    }
}

extern "C" void kernel_launch(void* const* d_in, const int* in_sizes, int n_in,
                              void* d_out, int out_size, void* d_ws, size_t ws_size,
                              hipStream_t stream) {
    const float* Xz      = (const float*)d_in[0];
    const float* centers = (const float*)d_in[1];
    const float* sigmas  = (const float*)d_in[2];
    const float* theta   = (const float*)d_in[3];
    float* out = (float*)d_out;
    float* y   = out;                             // N
    float* w   = out + NROWS;                     // N*32
    float* phi = out + (size_t)NROWS * 33;        // N*2080
    tsk_wmma_kernel<<<NROWS / TM, 256, 0, stream>>>(Xz, centers, sigmas, theta, y, w, phi);
}